// DynamicGraphConstructor_51204600103135
// MI455X (gfx1250) — compile-verified
//
#include <hip/hip_runtime.h>

#define NB 32
#define NC 64
#define NT 1000
#define NF 16
#define ND (NT * NF)          // 16000 elements per (b,c)
#define KT 128                // k-tile width staged in LDS
#define NKT (ND / KT)         // 125 k-tiles per batch
#define KCHUNKS 25            // blocks per batch along K
#define TPB (NKT / KCHUNKS)   // 5 k-tiles per block
#define LSTRIDE 132           // LDS row stride in floats (132 mod 64 == 4 -> conflict free)

typedef __attribute__((ext_vector_type(2))) float v2f;
typedef __attribute__((ext_vector_type(8))) float v8f;

#if defined(__gfx1250__) && __has_builtin(__builtin_amdgcn_wmma_f32_16x16x4_f32)
#define HAVE_WMMA_F32 1
#else
#define HAVE_WMMA_F32 0
#endif

// ---------------------------------------------------------------------------
// Zero the workspace (gram + row sums). Harness poisons ws, so we must clear.
// ---------------------------------------------------------------------------
__global__ __launch_bounds__(256) void zero_kernel(float* __restrict__ p, int n) {
    int i = blockIdx.x * 256 + threadIdx.x;
    if (i < n) p[i] = 0.0f;
}

// ---------------------------------------------------------------------------
// Fused raw-Gram (X * X^T per batch, fp32 WMMA) + row sums.
// grid = (KCHUNKS, NB), block = 256 (8 waves of 32).
// Each block: 5 k-tiles of 128 columns (partition of the 125 tiles),
//   stages 64x128 fp32 into LDS, waves compute 16x16 WMMA tiles,
//   threads 0..63 accumulate row sums from the staged LDS data.
// ---------------------------------------------------------------------------
__global__ __launch_bounds__(256)
void gram_kernel(const float* __restrict__ x,
                 float* __restrict__ gram,   // [NB][NC][NC]
                 float* __restrict__ S)      // [NB][NC] row sums
{
    __shared__ float lds[64 * LSTRIDE];

    const int kb   = blockIdx.x;      // 0..KCHUNKS-1
    const int b    = blockIdx.y;      // 0..NB-1
    const int tid  = threadIdx.x;
    const int lane = tid & 31;
    const int wave = tid >> 5;

    const float* __restrict__ xb = x + (size_t)b * NC * ND;

    // Wave -> output tiles: wave w owns (ti = w>>1, tj0 = (w&1)*2) and tj0+1.
    const int ti  = wave >> 1;
    const int tj0 = (wave & 1) * 2;

    // Fragment lane addressing (ISA 7.12.2):
    //   A 16x4 f32 : lane l -> row M = l&15, VGPR v holds K = (l>>4)*2 + v
    //   B 4x16 f32 : lane l -> col N = l&15, VGPR v holds K = (l>>4)*2 + v
    const int m   = lane & 15;
    const int kp  = (lane >> 4) * 2;

    v8f acc0 = {0.f, 0.f, 0.f, 0.f, 0.f, 0.f, 0.f, 0.f};
    v8f acc1 = {0.f, 0.f, 0.f, 0.f, 0.f, 0.f, 0.f, 0.f};
#if !HAVE_WMMA_F32
    float facc0[8] = {0}, facc1[8] = {0};
#endif

    float rowsum = 0.0f;   // valid for tid < 64

    for (int t = 0; t < TPB; ++t) {
        const int ktile = kb + t * KCHUNKS;   // 0..124, partition across blocks
        const int k0g   = ktile * KT;

        // ---- stage 64 x 128 fp32 tile into LDS (coalesced b128 loads) ----
#pragma unroll
        for (int i = 0; i < 8; ++i) {
            const int idx = tid + i * 256;        // 0..2047 float4 slots
            const int row = idx >> 5;             // 0..63
            const int c4  = idx & 31;             // 0..31 float4 within row
            const float4 v =
                *reinterpret_cast<const float4*>(xb + (size_t)row * ND + k0g + c4 * 4);
            *reinterpret_cast<float4*>(&lds[row * LSTRIDE + c4 * 4]) = v;
        }
        __syncthreads();

        // ---- row sums from the staged data (waves 0..1, uniform branch) ----
        if (tid < 64) {
            const float* rp = &lds[tid * LSTRIDE];
            float s = 0.0f;
#pragma unroll 8
            for (int k = 0; k < KT; ++k) s += rp[k];
            rowsum += s;
        }

        // ---- WMMA: 32 K=4 steps over this tile, 2 output tiles per wave ----
        const float* aRow  = &lds[(ti * 16 + m) * LSTRIDE];
        const float* bRow0 = &lds[(tj0 * 16 + m) * LSTRIDE];
        const float* bRow1 = &lds[((tj0 + 1) * 16 + m) * LSTRIDE];
#pragma unroll
        for (int ks = 0; ks < KT; ks += 4) {
            const v2f a  = *reinterpret_cast<const v2f*>(aRow  + ks + kp);
            const v2f b0 = *reinterpret_cast<const v2f*>(bRow0 + ks + kp);
            const v2f b1 = *reinterpret_cast<const v2f*>(bRow1 + ks + kp);
#if HAVE_WMMA_F32
            acc0 = __builtin_amdgcn_wmma_f32_16x16x4_f32(
                       false, a, false, b0, (short)0, acc0, false, false);
            acc1 = __builtin_amdgcn_wmma_f32_16x16x4_f32(
                       false, a, false, b1, (short)0, acc1, false, false);
#else
            // Scalar fallback (layout-inexact, keeps file compiling if the
            // f32 WMMA builtin signature differs; replaced on next round).
            for (int v = 0; v < 8; ++v) {
                facc0[v] += a[0] * b0[0] + a[1] * b0[1];
                facc1[v] += a[0] * b1[0] + a[1] * b1[1];
            }
#endif
        }
        __syncthreads();
    }

#if !HAVE_WMMA_F32
    for (int v = 0; v < 8; ++v) { acc0[v] = facc0[v]; acc1[v] = facc1[v]; }
#endif

    // ---- commit: D 16x16 f32 layout: lane l, VGPR v -> M = v + (l>>4)*8, N = l&15
    const int n   = lane & 15;
    const int mhi = (lane >> 4) * 8;
    float* __restrict__ gb = gram + (size_t)b * NC * NC;
#pragma unroll
    for (int v = 0; v < 8; ++v) {
        const int r  = ti * 16 + mhi + v;
        const int c0 = tj0 * 16 + n;
        atomicAdd(&gb[(size_t)r * NC + c0],      acc0[v]);
        atomicAdd(&gb[(size_t)r * NC + c0 + 16], acc1[v]);
    }
    if (tid < 64) {
        atomicAdd(&S[b * NC + tid], rowsum);
    }
}

// ---------------------------------------------------------------------------
// Finalize: mean/std from S + diag(G), normalized correlation, threshold,
// identity, symmetric degree normalization. One block per batch.
// ---------------------------------------------------------------------------
__global__ __launch_bounds__(256)
void finalize_kernel(const float* __restrict__ gram,
                     const float* __restrict__ S,
                     float* __restrict__ out)
{
    __shared__ float A[NC * NC];
    __shared__ float meanS[NC];
    __shared__ float invStd[NC];
    __shared__ float deg[NC];
    __shared__ float dsi[NC];

    const int b   = blockIdx.x;
    const int tid = threadIdx.x;
    const float* __restrict__ G  = gram + (size_t)b * NC * NC;
    const float* __restrict__ Sb = S + b * NC;

    if (tid < NC) {
        const float mval = Sb[tid] * (1.0f / (float)ND);
        const float diag = G[tid * NC + tid];
        float var = (diag - (float)ND * mval * mval) * (1.0f / (float)(ND - 1));
        var = fmaxf(var, 0.0f);
        meanS[tid]  = mval;
        invStd[tid] = 1.0f / (sqrtf(var) + 1e-8f);
        deg[tid]    = 0.0f;
    }
    __syncthreads();

    for (int idx = tid; idx < NC * NC; idx += 256) {
        const int r = idx >> 6;
        const int c = idx & 63;
        // Sum of products of normalized rows = (G_rc - D*m_r*m_c)/(s_r*s_c)
        const float gn = (G[idx] - (float)ND * meanS[r] * meanS[c]) * invStd[r] * invStd[c];
        const float adjv = gn * (1.0f / (float)NT);
        float a = (adjv > 0.5f) ? 1.0f : 0.0f;
        if (r == c) a += 1.0f;
        A[idx] = a;
        atomicAdd(&deg[r], a);   // ds_add_f32
    }
    __syncthreads();

    if (tid < NC) dsi[tid] = rsqrtf(deg[tid] + 1e-8f);
    __syncthreads();

    float* __restrict__ ob = out + (size_t)b * NC * NC;
    for (int idx = tid; idx < NC * NC; idx += 256) {
        const int r = idx >> 6;
        const int c = idx & 63;
        ob[idx] = A[idx] * dsi[r] * dsi[c];
    }
}

// ---------------------------------------------------------------------------
extern "C" void kernel_launch(void* const* d_in, const int* in_sizes, int n_in,
                              void* d_out, int out_size, void* d_ws, size_t ws_size,
                              hipStream_t stream) {
    const float* x   = (const float*)d_in[0];
    float*       out = (float*)d_out;

    float* gram = (float*)d_ws;                 // [NB*NC*NC] fp32
    float* S    = gram + (size_t)NB * NC * NC;  // [NB*NC]    fp32

    const int nzero = NB * NC * NC + NB * NC;
    zero_kernel<<<(nzero + 255) / 256, 256, 0, stream>>>(gram, nzero);

    gram_kernel<<<dim3(KCHUNKS, NB), 256, 0, stream>>>(x, gram, S);

    finalize_kernel<<<NB, 256, 0, stream>>>(gram, S, out);
}